// Linear_59064390255278
// MI455X (gfx1250) — compile-verified
//
#include <hip/hip_runtime.h>
#include <hip/hip_bf16.h>

// CDNA5 / gfx1250: wave32, WMMA f32 16x16x4 path.
typedef __attribute__((ext_vector_type(2))) float v2f;
typedef __attribute__((ext_vector_type(4))) float v4f;
typedef __attribute__((ext_vector_type(8))) float v8f;

// ---------------------------------------------------------------------------
// Weight repack: W[u*MUL + w] * scale  ->  WMMA-B fragment order, so the GEMM
// kernel fetches one b128 per lane per N-tile covering TWO K-steps (k0, k0+4).
//
// packed[((k8*NT + t)*32 + lane)*4 + q] = scale * W[K(q)][t*16 + (lane&15)]
//   K(q) = k8*8 + (q>>1)*4 + (lane>>4)*2 + (q&1)
// ---------------------------------------------------------------------------
template <int MUL>
__global__ __launch_bounds__(256)
void repack_w_kernel(const float* __restrict__ w, float* __restrict__ pw,
                     float scale)
{
    constexpr int NT = MUL / 16;
    const int p = blockIdx.x * 256 + threadIdx.x;     // [0, MUL*MUL)
    const int q    = p & 3;
    const int lane = (p >> 2) & 31;
    const int t    = (p >> 7) % NT;
    const int k8   = (p >> 7) / NT;
    const int K = k8 * 8 + (q >> 1) * 4 + (lane >> 4) * 2 + (q & 1);
    const int N = t * 16 + (lane & 15);
    pw[p] = w[(size_t)K * MUL + N] * scale;
}

// ---------------------------------------------------------------------------
// Packed-B GEMM. One block = 8 waves = 128 batch rows; each wave owns a
// 16-row slab and the FULL N panel (MUL/16 accumulator tiles) so x is read
// exactly once from HBM. Scale is pre-folded into pw; bias pre-loaded into C.
// Inner step (k8) = 2 K-steps: 16x global_load_b128 (B) + 2x b64/4x b32 (A)
// + 32 WMMAs.
// ---------------------------------------------------------------------------
template <int MUL, int D, bool HAS_BIAS>
__global__ __launch_bounds__(256, 2)
void eqlin_wmma_packed_kernel(const float* __restrict__ x,
                              const float* __restrict__ pw,
                              const float* __restrict__ bs,
                              float* __restrict__ out,
                              int seg_off)
{
    constexpr int DIM = 1664;
    constexpr int NT  = MUL / 16;

    const int wave = threadIdx.x >> 5;
    const int lane = threadIdx.x & 31;
    const int half = lane >> 4;
    const int lm   = lane & 15;

    const int i  = blockIdx.y;                       // 0..D-1
    const int b0 = blockIdx.x * 128 + wave * 16;     // batch-row base

    const float* __restrict__ xrow = x + (size_t)(b0 + lm) * DIM + seg_off + i;
    const float* __restrict__ pwl  = pw + lane * 4;  // lane's packed-B stream

    // C-matrix init: bias broadcast along M (same bias for every row r).
    v8f acc[NT];
    #pragma unroll
    for (int t = 0; t < NT; ++t) {
        const float binit = HAS_BIAS ? bs[t * 16 + lm] : 0.0f;
        #pragma unroll
        for (int r = 0; r < 8; ++r) acc[t][r] = binit;
    }

    #pragma unroll 2
    for (int k8 = 0; k8 < MUL / 8; ++k8) {
        const int ka = k8 * 8 + 2 * half;
        // A frags for the two K-steps (u-stride = D floats; D==1 merges to b64)
        v2f a0, a1;
        a0.x = xrow[(size_t)(ka + 0) * D];
        a0.y = xrow[(size_t)(ka + 1) * D];
        a1.x = xrow[(size_t)(ka + 4) * D];
        a1.y = xrow[(size_t)(ka + 5) * D];

        #pragma unroll
        for (int t = 0; t < NT; ++t) {
            const v4f b = *(const v4f*)(pwl + (size_t)(k8 * NT + t) * 128);
            const v2f blo = {b.x, b.y};
            const v2f bhi = {b.z, b.w};
            acc[t] = __builtin_amdgcn_wmma_f32_16x16x4_f32(
                false, a0, false, blo, (short)0, acc[t], false, false);
            acc[t] = __builtin_amdgcn_wmma_f32_16x16x4_f32(
                false, a1, false, bhi, (short)0, acc[t], false, false);
        }
    }

    // C/D layout: VGPR r -> M = half*8 + r, N = lm (within tile t).
    #pragma unroll
    for (int t = 0; t < NT; ++t) {
        float* __restrict__ orow = out + seg_off + (size_t)(t * 16 + lm) * D + i;
        #pragma unroll
        for (int r = 0; r < 8; ++r) {
            const int m = half * 8 + r;
            orow[(size_t)(b0 + m) * DIM] = acc[t][r];
        }
    }
}

// ---------------------------------------------------------------------------
// Fallback (no scratch): direct-load variant from round 1.
// ---------------------------------------------------------------------------
template <int MUL, int D, bool HAS_BIAS>
__global__ __launch_bounds__(256, 2)
void eqlin_wmma_kernel(const float* __restrict__ x,
                       const float* __restrict__ w,
                       const float* __restrict__ bs,
                       float* __restrict__ out,
                       int seg_off, float scale)
{
    constexpr int DIM = 1664;
    constexpr int NT  = MUL / 16;

    const int wave = threadIdx.x >> 5;
    const int lane = threadIdx.x & 31;
    const int half = lane >> 4;
    const int lm   = lane & 15;

    const int i  = blockIdx.y;
    const int b0 = blockIdx.x * 128 + wave * 16;

    const float* __restrict__ xrow = x + (size_t)(b0 + lm) * DIM + seg_off + i;

    v8f acc[NT] = {};

    #pragma unroll 2
    for (int k0 = 0; k0 < MUL; k0 += 4) {
        const int ka = k0 + 2 * half;
        v2f a;
        a.x = xrow[(size_t)(ka + 0) * D];
        a.y = xrow[(size_t)(ka + 1) * D];
        #pragma unroll
        for (int t = 0; t < NT; ++t) {
            v2f b;
            b.x = w[(size_t)(ka + 0) * MUL + t * 16 + lm];
            b.y = w[(size_t)(ka + 1) * MUL + t * 16 + lm];
            acc[t] = __builtin_amdgcn_wmma_f32_16x16x4_f32(
                false, a, false, b, (short)0, acc[t], false, false);
        }
    }

    #pragma unroll
    for (int t = 0; t < NT; ++t) {
        const int n = t * 16 + lm;
        const float bias = HAS_BIAS ? bs[n] : 0.0f;
        float* __restrict__ orow = out + seg_off + (size_t)n * D + i;
        #pragma unroll
        for (int r = 0; r < 8; ++r) {
            const int m = half * 8 + r;
            orow[(size_t)(b0 + m) * DIM] = acc[t][r] * scale + bias;
        }
    }
}

extern "C" void kernel_launch(void* const* d_in, const int* in_sizes, int n_in,
                              void* d_out, int out_size, void* d_ws, size_t ws_size,
                              hipStream_t stream) {
    (void)in_sizes; (void)n_in; (void)out_size;

    const float* ws = (const float*)d_in[0];   // 147456 weights (3 blocks)
    const float* bs = (const float*)d_in[1];   // 256 biases (scalar block only)
    const float* x  = (const float*)d_in[2];   // [32768, 1664]
    float*       out = (float*)d_out;          // [32768, 1664]

    constexpr int BATCH = 32768;
    constexpr float S256 = 1.0f / 16.0f;               // 1/sqrt(256)
    constexpr float S128 = 0.08838834764831845f;       // 1/sqrt(128)
    const dim3 block(256);
    const int row_tiles = BATCH / 128;                 // 256 blocks along batch

    const size_t need = (size_t)147456 * sizeof(float);
    if (ws_size >= need) {
        float* pw = (float*)d_ws;
        // Repack weights (scale folded in): seg offsets match source offsets.
        repack_w_kernel<256><<<256 * 256 / 256, block, 0, stream>>>(ws,          pw,          S256);
        repack_w_kernel<256><<<256 * 256 / 256, block, 0, stream>>>(ws + 65536,  pw + 65536,  S256);
        repack_w_kernel<128><<<128 * 128 / 256, block, 0, stream>>>(ws + 131072, pw + 131072, S128);

        eqlin_wmma_packed_kernel<256, 1, true>
            <<<dim3(row_tiles, 1), block, 0, stream>>>(x, pw,          bs,      out, 0);
        eqlin_wmma_packed_kernel<256, 3, false>
            <<<dim3(row_tiles, 3), block, 0, stream>>>(x, pw + 65536,  nullptr, out, 256);
        eqlin_wmma_packed_kernel<128, 5, false>
            <<<dim3(row_tiles, 5), block, 0, stream>>>(x, pw + 131072, nullptr, out, 1024);
    } else {
        eqlin_wmma_kernel<256, 1, true>
            <<<dim3(row_tiles, 1), block, 0, stream>>>(x, ws,          bs,      out, 0,    S256);
        eqlin_wmma_kernel<256, 3, false>
            <<<dim3(row_tiles, 3), block, 0, stream>>>(x, ws + 65536,  nullptr, out, 256,  S256);
        eqlin_wmma_kernel<128, 5, false>
            <<<dim3(row_tiles, 5), block, 0, stream>>>(x, ws + 131072, nullptr, out, 1024, S128);
    }
}